// EuclidToHypAlign_61727269978358
// MI455X (gfx1250) — compile-verified
//
#include <hip/hip_runtime.h>
#include <hip/hip_bf16.h>
#include <math.h>
#include <stdint.h>

// Problem constants (from reference)
#define BB 4
#define AA 16
#define NN 8
#define LL 512
#define DE 256
#define DH 64
#define HH 8
#define DHEAD 32
#define NL (NN*LL)          // 4096
#define LN_EPS 1e-5f

typedef __attribute__((ext_vector_type(2))) float v2f;
typedef __attribute__((ext_vector_type(8))) float v8f;
typedef __attribute__((ext_vector_type(4))) unsigned int v4u;
typedef __attribute__((ext_vector_type(4))) int v4i;
typedef __attribute__((ext_vector_type(8))) int v8i;

// ---------------------------------------------------------------------------
// Kernel A: per (b,a): LN(curr_rho) -> q = xn @ Wq^T -> fold Wk into qW.
// qW stored pair-interleaved: qW[(b*A+a)*4096 + (e>>1)*32 + h*2 + (e&1)],
// heads 8..15 zero-padded, scaled by 1/sqrt(DHEAD).
// ---------------------------------------------------------------------------
__global__ void kA_qw(const float* __restrict__ curr_rho,
                      const float* __restrict__ Wq,
                      const float* __restrict__ Wk,
                      const float* __restrict__ lnw,
                      const float* __restrict__ lnb,
                      float* __restrict__ qW_ws) {
    __shared__ float xn[DE];
    __shared__ float qs[DE];
    __shared__ float red[256];
    const int ba = blockIdx.x;          // b*A + a
    const int t  = threadIdx.x;         // 0..255 (== DE)

    float x = curr_rho[(size_t)ba * DE + t];
    red[t] = x; __syncthreads();
    for (int s = 128; s > 0; s >>= 1) { if (t < s) red[t] += red[t + s]; __syncthreads(); }
    float mu = red[0] * (1.0f / DE);
    __syncthreads();
    float d = x - mu;
    red[t] = d * d; __syncthreads();
    for (int s = 128; s > 0; s >>= 1) { if (t < s) red[t] += red[t + s]; __syncthreads(); }
    float rstd = rsqrtf(red[0] * (1.0f / DE) + LN_EPS);
    xn[t] = d * rstd * lnw[t] + lnb[t];
    __syncthreads();

    // q[t] = dot(xn, Wq[t,:])
    {
        const float* wr = Wq + (size_t)t * DE;
        float acc = 0.f;
        #pragma unroll 8
        for (int e = 0; e < DE; ++e) acc += xn[e] * wr[e];
        qs[t] = acc;
    }
    __syncthreads();

    // qW[h][e] = (1/sqrt(32)) * sum_d q[h*32+d] * Wk[h*32+d][e]
    const float scale = 0.17677669529663688f; // 1/sqrt(32)
    const int e = t;
    float outh[HH];
    for (int h = 0; h < HH; ++h) {
        float s = 0.f;
        const float* qh = qs + h * DHEAD;
        #pragma unroll 8
        for (int dd = 0; dd < DHEAD; ++dd)
            s += qh[dd] * Wk[(size_t)(h * DHEAD + dd) * DE + e];
        outh[h] = s * scale;
    }
    float* dst = qW_ws + (size_t)ba * (DE * 16);
    const int base = (e >> 1) * 32 + (e & 1);
    for (int h = 0; h < HH; ++h) dst[base + h * 2] = outh[h];
    for (int h = HH; h < 16; ++h) dst[base + h * 2] = 0.f;
}

// ---------------------------------------------------------------------------
// Kernel B: streaming scores with TDM staging.
// 4 waves/block; each wave TDM-loads its 16x256 demo_rho tile (row stride
// 16KB in memory) into LDS with 2-DWORD row padding (bank-conflict free),
// waits on TENSORcnt, computes LN stats via wave shuffles, then runs
// 64x v_wmma_f32_16x16x4_f32 with A built from the LDS tile and B from qW.
// Writes raw scores into the attn region of d_out.
// ---------------------------------------------------------------------------
#define WAVES_B  4
#define TILE_F   258                              // padded LDS row stride (floats)
#define SQW_OFF  0                                // 4096 floats
#define SW_OFF   4096
#define SB_OFF   4352
#define STAT_OFF 4608                             // [wave][mean16|rstd16]
#define TILE_OFF 4736
#define TILE_WAVE (16 * TILE_F)                   // 4128 floats
#define SMEM_FLOATS (TILE_OFF + WAVES_B * TILE_WAVE)   // 21248 floats (~85KB)

__global__ void kB_scores(const float* __restrict__ demo_rho,
                          const float* __restrict__ qW_ws,
                          const float* __restrict__ lnkw,
                          const float* __restrict__ lnkb,
                          float* __restrict__ attn_out) {
    extern __shared__ float smem[];
    const int t = threadIdx.x;
    const int wave = t >> 5, lane = t & 31;
    const int tilesPerBA = NL / 16 / WAVES_B;      // 64
    const int ba = blockIdx.x / tilesPerBA;
    const int tg = blockIdx.x % tilesPerBA;
    const int b = ba >> 4, a = ba & 15;
    const int x0 = (tg * WAVES_B + wave) * 16;

    // --- TDM: DMA this wave's 16x256 tile into its padded LDS buffer -------
    {
        const float* gsrc = demo_rho + ((size_t)(b * NL + x0) * AA + a) * DE;
        uint64_t gaddr = (uint64_t)(uintptr_t)gsrc;
        uint32_t ldsb = (uint32_t)__builtin_amdgcn_readfirstlane(
                            (int)((TILE_OFF + wave * TILE_WAVE) * 4));
        uint32_t glo = (uint32_t)__builtin_amdgcn_readfirstlane((int)(uint32_t)(gaddr & 0xffffffffull));
        uint32_t ghi = (uint32_t)__builtin_amdgcn_readfirstlane((int)(uint32_t)(gaddr >> 32));
        // group0: count=1 | lds_addr | global_addr[56:0] | type=2
        v4u g0 = { 1u, ldsb, glo, (ghi & 0x01FFFFFFu) | 0x80000000u };
        // group1: data_size=4B, pad_enable, pad_interval=7 (256 dw), pad_amount=1 (2 dw),
        //         tensor_dim0=256, tensor_dim1=16, tile_dim0=256, tile_dim1=16,
        //         tensor_dim0_stride=4096 elements
        v8i g1 = { (int)0x03D20000u,        // mask=0 | ds=2<<16 | pad_en | intv=7<<22 | amt=1<<25
                   (int)(256u << 16),       // tensor_dim0[15:0] in [63:48]
                   (int)(16u  << 16),       // tensor_dim1[15:0] in [95:80]
                   (int)(256u << 16),       // tile_dim0 in [127:112]
                   16,                      // tile_dim1 in [143:128]
                   4096,                    // tensor_dim0_stride[31:0]
                   0, 0 };
        v4i gz = { 0, 0, 0, 0 };
#if __clang_major__ >= 23
        v8i gz8 = { 0, 0, 0, 0, 0, 0, 0, 0 };
        __builtin_amdgcn_tensor_load_to_lds(g0, g1, gz, gz, gz8, 0);
#else
        __builtin_amdgcn_tensor_load_to_lds(g0, g1, gz, gz, 0);
#endif
    }

    // cooperative fill of qW + LN params while the TDM runs
    const float* qsrc = qW_ws + (size_t)ba * (DE * 16);
    for (int i = t; i < DE * 16; i += blockDim.x) smem[SQW_OFF + i] = qsrc[i];
    for (int i = t; i < DE; i += blockDim.x) { smem[SW_OFF + i] = lnkw[i]; smem[SB_OFF + i] = lnkb[i]; }

    __builtin_amdgcn_s_wait_tensorcnt(0);
    __syncthreads();

    float* tile = smem + TILE_OFF + wave * TILE_WAVE;

    // pass 1: LN stats for the wave's 16 rows (from LDS)
    for (int r = 0; r < 16; ++r) {
        const float* rp = tile + r * TILE_F;
        float s = 0.f, sq = 0.f;
        #pragma unroll
        for (int e = lane; e < DE; e += 32) { float v = rp[e]; s += v; sq += v * v; }
        for (int off = 16; off >= 1; off >>= 1) {
            s  += __shfl_xor(s,  off);
            sq += __shfl_xor(sq, off);
        }
        if (lane == 0) {
            float mu  = s * (1.0f / DE);
            float var = sq * (1.0f / DE) - mu * mu;
            smem[STAT_OFF + wave * 32 + r]      = mu;
            smem[STAT_OFF + wave * 32 + 16 + r] = rsqrtf(var + LN_EPS);
        }
    }
    __syncthreads();

    // pass 2: WMMA K-loop (A built on the fly from the LN'd LDS tile)
    const int m = lane & 15;
    const int half = lane >> 4;                    // 0 or 1
    const float mu = smem[STAT_OFF + wave * 32 + m];
    const float rs = smem[STAT_OFF + wave * 32 + 16 + m];
    const float* rowp = tile + m * TILE_F;
    const int n = m;                               // output column (head, padded)

    v8f c = {0.f,0.f,0.f,0.f,0.f,0.f,0.f,0.f};
    #pragma unroll 4
    for (int k4 = 0; k4 < DE; k4 += 4) {
        const int kk = k4 + half * 2;
        float2 xv = *reinterpret_cast<const float2*>(rowp + kk);
        v2f av; av.x = (xv.x - mu) * rs * smem[SW_OFF + kk]     + smem[SB_OFF + kk];
                av.y = (xv.y - mu) * rs * smem[SW_OFF + kk + 1] + smem[SB_OFF + kk + 1];
        float2 bv = *reinterpret_cast<const float2*>(&smem[SQW_OFF + (kk >> 1) * 32 + n * 2]);
        v2f bf; bf.x = bv.x; bf.y = bv.y;
        c = __builtin_amdgcn_wmma_f32_16x16x4_f32(false, av, false, bf,
                                                  (short)0, c, false, false);
    }

    // D tile: VGPR r -> (M = r + half*8, N = n). Rows are contiguous in x.
    if (n < HH) {
        float* op = attn_out + (((size_t)b * HH + n) * AA + a) * NL + x0 + half * 8;
        *reinterpret_cast<float4*>(op)     = make_float4(c[0], c[1], c[2], c[3]);
        *reinterpret_cast<float4*>(op + 4) = make_float4(c[4], c[5], c[6], c[7]);
    }
}

// ---------------------------------------------------------------------------
// Softmax over N*L per (b,h,a), with mask -> -inf, in place on attn region.
// ---------------------------------------------------------------------------
__global__ void kSoftmax(float* __restrict__ attn, const int* __restrict__ mask) {
    __shared__ float row[NL];                       // 16KB
    __shared__ float red[256];
    const int t = threadIdx.x;
    const int r = blockIdx.x;                       // (b*H + h)*A + a
    const int b = r / (HH * AA);
    float* p = attn + (size_t)r * NL;
    const int* mb = mask + (size_t)b * NL;

    float mx = -INFINITY;
    for (int i = t; i < NL; i += 256) {
        float v = p[i];
        if (mb[i] == 0) v = -INFINITY;
        row[i] = v;
        mx = fmaxf(mx, v);
    }
    red[t] = mx; __syncthreads();
    for (int s = 128; s > 0; s >>= 1) { if (t < s) red[t] = fmaxf(red[t], red[t + s]); __syncthreads(); }
    mx = red[0]; __syncthreads();

    float sum = 0.f;
    for (int i = t; i < NL; i += 256) { float e = expf(row[i] - mx); row[i] = e; sum += e; }
    red[t] = sum; __syncthreads();
    for (int s = 128; s > 0; s >>= 1) { if (t < s) red[t] += red[t + s]; __syncthreads(); }
    const float inv = 1.f / red[0];
    for (int i = t; i < NL; i += 256) p[i] = row[i] * inv;
}

// ---------------------------------------------------------------------------
// Kernel C: vals = logmap0(demo_hyp), wave per row of 64 (2 elems/lane).
// ---------------------------------------------------------------------------
__global__ void kLogmap(const float* __restrict__ hyp, float* __restrict__ vals) {
    const int t = threadIdx.x;
    const int wave = t >> 5, lane = t & 31;
    const size_t row = (size_t)blockIdx.x * 8 + wave;     // < B*NL
    const float* p = hyp + row * DH;
    float v0 = p[lane], v1 = p[lane + 32];
    float n2 = v0 * v0 + v1 * v1;
    for (int off = 16; off >= 1; off >>= 1) n2 += __shfl_xor(n2, off);
    float n  = fmaxf(sqrtf(n2), 1e-15f);
    float u  = fminf(n, 1.f - 1e-5f);
    float at = 0.5f * logf((1.f + u) / (1.f - u));        // arctanh(u)
    float s  = at / n;
    float* o = vals + row * DH;
    o[lane] = v0 * s; o[lane + 32] = v1 * s;
}

// ---------------------------------------------------------------------------
// Kernel D: m_h = attn @ vals via v_wmma_f32_16x16x4_f32.
// grid = B*8 (m-tiles), block = 128 (4 waves = 4 n-tiles of 16).
// ---------------------------------------------------------------------------
__global__ void kD_mh(const float* __restrict__ attn,
                      const float* __restrict__ vals,
                      float* __restrict__ mh) {
    const int t = threadIdx.x;
    const int wave = t >> 5, lane = t & 31;
    const int b  = blockIdx.x >> 3;
    const int mt = blockIdx.x & 7;
    const int m = lane & 15, half = lane >> 4;
    const int n = wave * 16 + m;                    // 0..63
    const float* ap = attn + ((size_t)b * HH * AA + mt * 16 + m) * NL;
    const float* vp = vals + (size_t)b * NL * DH + n;

    v8f c = {0.f,0.f,0.f,0.f,0.f,0.f,0.f,0.f};
    #pragma unroll 4
    for (int k4 = 0; k4 < NL; k4 += 4) {
        const int kk = k4 + half * 2;
        float2 av2 = *reinterpret_cast<const float2*>(ap + kk);
        v2f av; av.x = av2.x; av.y = av2.y;
        v2f bf; bf.x = vp[(size_t)kk * DH]; bf.y = vp[(size_t)(kk + 1) * DH];
        c = __builtin_amdgcn_wmma_f32_16x16x4_f32(false, av, false, bf,
                                                  (short)0, c, false, false);
    }
    float* op = mh + ((size_t)b * HH * AA + mt * 16 + half * 8) * DH + n;
    #pragma unroll
    for (int r = 0; r < 8; ++r) op[(size_t)r * DH] = c[r];
}

// ---------------------------------------------------------------------------
// Kernel E: hyperbolic tail. One block per b.
// y = logmap0(expmap0(m_h)); mean over h; expmap0; radius clamp; mean over a.
// ---------------------------------------------------------------------------
__global__ void kE_final(const float* __restrict__ mh, float* __restrict__ outp) {
    __shared__ float accA[AA][DH];                  // 4KB
    __shared__ float accB[DH];
    const int t = threadIdx.x;
    const int wave = t >> 5, lane = t & 31;
    const int b = blockIdx.x;
    for (int i = t; i < AA * DH; i += 256) (&accA[0][0])[i] = 0.f;
    if (t < DH) accB[t] = 0.f;
    __syncthreads();

    const float maxrad = 1.f - 1e-5f;
    for (int row = wave; row < HH * AA; row += 8) {
        const int a = row & 15;                     // row = h*A + a
        const float* p = mh + ((size_t)b * HH * AA + row) * DH;
        float v0 = p[lane], v1 = p[lane + 32];
        float n2 = v0 * v0 + v1 * v1;
        for (int off = 16; off >= 1; off >>= 1) n2 += __shfl_xor(n2, off);
        float sn = sqrtf(n2);
        float n  = fmaxf(sn, 1e-15f);
        float th = tanhf(n);
        float w0 = th * v0 / n, w1 = th * v1 / n;   // expmap0
        float nw = fmaxf(th * sn / n, 1e-15f);      // ||w||
        float u  = fminf(nw, maxrad);
        float at = 0.5f * logf((1.f + u) / (1.f - u));
        float s  = (at / nw) * 0.125f;              // fold mean over H
        atomicAdd(&accA[a][lane],      s * w0);
        atomicAdd(&accA[a][lane + 32], s * w1);
    }
    __syncthreads();

    for (int a = wave; a < AA; a += 8) {
        float v0 = accA[a][lane], v1 = accA[a][lane + 32];
        float n2 = v0 * v0 + v1 * v1;
        for (int off = 16; off >= 1; off >>= 1) n2 += __shfl_xor(n2, off);
        float sn = sqrtf(n2);
        float n  = fmaxf(sn, 1e-15f);
        float th = tanhf(n);
        float e0 = th * v0 / n, e1 = th * v1 / n;   // expmap0
        float nrm = fmaxf(th * sn / n, 1e-6f);
        float sc  = fminf(maxrad / nrm, 1.f);
        atomicAdd(&accB[lane],      e0 * sc * (1.f / AA));
        atomicAdd(&accB[lane + 32], e1 * sc * (1.f / AA));
    }
    __syncthreads();
    if (t < DH) outp[(size_t)b * DH + t] = accB[t];
}

// ---------------------------------------------------------------------------
extern "C" void kernel_launch(void* const* d_in, const int* in_sizes, int n_in,
                              void* d_out, int out_size, void* d_ws, size_t ws_size,
                              hipStream_t stream) {
    (void)in_sizes; (void)n_in; (void)out_size; (void)ws_size;
    const float* curr_rho = (const float*)d_in[0];
    const float* demo_rho = (const float*)d_in[1];
    const float* demo_hyp = (const float*)d_in[2];
    const int*   demo_mask= (const int*)  d_in[3];
    const float* Wq       = (const float*)d_in[4];
    const float* Wk       = (const float*)d_in[5];
    const float* ln_q_w   = (const float*)d_in[6];
    const float* ln_q_b   = (const float*)d_in[7];
    const float* ln_k_w   = (const float*)d_in[8];
    const float* ln_k_b   = (const float*)d_in[9];

    float* out      = (float*)d_out;
    float* curr_hyp = out;                      // B*DH = 256 floats
    float* attn     = out + BB * DH;            // B*H*A*NL floats

    float* ws   = (float*)d_ws;
    float* qW   = ws;                                   // 64 * 4096   = 262144
    float* vals = qW + (size_t)BB * AA * DE * 16;       // B*NL*DH     = 1048576
    float* mh   = vals + (size_t)BB * NL * DH;          // B*H*A*DH    = 32768

    kA_qw    <<<BB * AA,                    256, 0, stream>>>(curr_rho, Wq, Wk, ln_q_w, ln_q_b, qW);
    kB_scores<<<BB * AA * (NL/16/WAVES_B),  WAVES_B * 32, SMEM_FLOATS * sizeof(float), stream>>>
             (demo_rho, qW, ln_k_w, ln_k_b, attn);
    kSoftmax <<<BB * HH * AA,               256, 0, stream>>>(attn, demo_mask);
    kLogmap  <<<BB * NL / 8,                256, 0, stream>>>(demo_hyp, vals);
    kD_mh    <<<BB * 8,                     128, 0, stream>>>(attn, vals, mh);
    kE_final <<<BB,                         256, 0, stream>>>(mh, curr_hyp);
}